// DefSeperateConv2d_76321568850100
// MI455X (gfx1250) — compile-verified
//
#include <hip/hip_runtime.h>
#include <math.h>

// Problem constants (from the reference)
#define B_    2
#define CIN_  64
#define COUT_ 64
#define H_    192
#define W_    192
#define HW_   (H_ * W_)
#define NPIX_ (B_ * HW_)       // 73728 = N of both GEMMs
#define G_    2
#define KK_   9
#define CG_   (CIN_ / G_)      // 32
#define KTOT_ (CIN_ * KK_)     // 576 = contraction length of both GEMMs
#define NCH27 27               // 18 semantic-offset + 9 modulation channels
#define NCHPAD 32              // padded to full M-tile height -> unconditional stores

#define TN_        64          // pixels per block in the GEMM kernels
#define LDS_STRIDE 65          // padded row stride (floats) for the S tile

typedef float v2f __attribute__((ext_vector_type(2)));
typedef float v8f __attribute__((ext_vector_type(8)));

__device__ __forceinline__ float sigmoid_(float x) {
    return 1.0f / (1.0f + __expf(-x));
}

__device__ __forceinline__ float samp_mask_(const float* __restrict__ p, int y, int x) {
    return (y >= 0 && y < H_ && x >= 0 && x < W_) ? p[y * W_ + x] : 0.0f;
}

// ---------------------------------------------------------------------------
// Kernel A: semantic-offset + modulation convs as one WMMA GEMM.
//   M = 32 (rows 0..17 = sem_w, 18..26 = m2_w, 27..31 = padding)
//   K = 576 (c*9 + tap),  N = 73728 pixels (64 per block).
// Writes conv result channel-major into a 32-channel padded workspace:
// ws_conv[ch * NPIX_ + p]; padding rows are stored but never read back.
// ---------------------------------------------------------------------------
__global__ void __launch_bounds__(256)
dsc_conv27_kernel(const float* __restrict__ input,
                  const float* __restrict__ sem_w,   // (18, 64, 3, 3) -> row-major (18, 576)
                  const float* __restrict__ m2_w,    // (9, 64, 3, 3)  -> row-major (9, 576)
                  float* __restrict__ ws_conv)       // (32, NPIX_) padded
{
    extern __shared__ float smem[];   // im2col tile: [KTOT_][LDS_STRIDE]
    const int pbase = blockIdx.x * TN_;

    // ---- Phase 1: stage zero-padded 3x3 im2col tile into LDS ----
    for (int task = threadIdx.x; task < CIN_ * TN_; task += 256) {
        const int n = task & (TN_ - 1);
        const int c = task >> 6;                  // TN_ == 64
        const int p  = pbase + n;
        const int b  = p / HW_;
        const int hw = p - b * HW_;
        const int y  = hw / W_;
        const int x  = hw - y * W_;
        const float* chan = input + (size_t)(b * CIN_ + c) * HW_;
#pragma unroll
        for (int t = 0; t < 9; ++t) {
            const int yy = y + t / 3 - 1;
            const int xx = x + t % 3 - 1;
            const bool inb = (yy >= 0) && (yy < H_) && (xx >= 0) && (xx < W_);
            const float v = inb ? chan[yy * W_ + xx] : 0.0f;
            smem[(c * 9 + t) * LDS_STRIDE + n] = v;
        }
    }
    __syncthreads();

    // ---- Phase 2: fp32 WMMA sweep, one 16x16 C-tile per wave ----
    const int lane   = threadIdx.x & 31;
    const int wv     = threadIdx.x >> 5;   // 0..7
    const int lane16 = lane & 15;
    const int half   = lane >> 4;
    const int koff   = half * 2;
    const int nt     = wv & 3;             // N tile
    const int mt     = wv >> 2;            // M tile (0..1)

    // weight row for this lane; rows >= 27 are padding -> clamp the *read*
    // pointer in-bounds; their results land in padding workspace rows.
    const int m = mt * 16 + lane16;        // 0..31
    const float* arow = (m < 18)
        ? (sem_w + (size_t)m * KTOT_)
        : (m2_w + (size_t)((m - 18) < 8 ? (m - 18) : 8) * KTOT_);
    const int bcol = nt * 16 + lane16;

    v8f acc = {0.f, 0.f, 0.f, 0.f, 0.f, 0.f, 0.f, 0.f};
#pragma unroll 4
    for (int k0 = 0; k0 < KTOT_; k0 += 4) {
        const v2f a = *(const v2f*)(arow + k0 + koff);
        const float* srow = smem + (k0 + koff) * LDS_STRIDE + bcol;
        const v2f bv = { srow[0], srow[LDS_STRIDE] };
        acc = __builtin_amdgcn_wmma_f32_16x16x4_f32(false, a, false, bv,
                                                    (short)0, acc, false, false);
    }

    // ---- Epilogue: unconditional channel-major coalesced stores ----
    const int p = pbase + nt * 16 + lane16;
#pragma unroll
    for (int j = 0; j < 8; ++j) {
        const int ch = mt * 16 + half * 8 + j;
        ws_conv[(size_t)ch * NPIX_ + p] = acc[j];
    }
}

// ---------------------------------------------------------------------------
// Kernel B: light per-pixel work: region/m1 convs (1 input channel),
// absolute sample coords, sigmoid modulation, bilinear update_mask.
// ---------------------------------------------------------------------------
__global__ void __launch_bounds__(256)
dsc_offsets_kernel(const float* __restrict__ mask_in,
                   const float* __restrict__ ws_conv,
                   const float* __restrict__ sem_b,  const float* __restrict__ reg_w,
                   const float* __restrict__ reg_b,  const float* __restrict__ m1_w,
                   const float* __restrict__ m1_b,   const float* __restrict__ m2_b,
                   float* __restrict__ ws_py, float* __restrict__ ws_px,
                   float* __restrict__ ws_dm, float* __restrict__ ws_um,
                   float* __restrict__ out_um)
{
    const int idx = blockIdx.x * 256 + threadIdx.x;
    if (idx >= NPIX_) return;
    const int b  = idx / HW_;
    const int hw = idx - b * HW_;
    const int y  = hw / W_;
    const int x  = hw - y * W_;

    // region + m1 convs over the 1-channel mask (cheap)
    float reg18[18]; float am1[9];
#pragma unroll
    for (int i = 0; i < 18; ++i) reg18[i] = 0.f;
#pragma unroll
    for (int i = 0; i < 9; ++i)  am1[i] = 0.f;
#pragma unroll
    for (int t = 0; t < 9; ++t) {
        const int yy = y + t / 3 - 1;
        const int xx = x + t % 3 - 1;
        const bool inb = (yy >= 0) && (yy < H_) && (xx >= 0) && (xx < W_);
        const float mv = inb ? mask_in[b * HW_ + yy * W_ + xx] : 0.0f;
#pragma unroll
        for (int ch = 0; ch < 18; ++ch)
            reg18[ch] = fmaf(reg_w[ch * 9 + t], mv, reg18[ch]);
#pragma unroll
        for (int ch = 0; ch < 9; ++ch)
            am1[ch] = fmaf(m1_w[ch * 9 + t], mv, am1[ch]);
    }

    // semantic conv results (WMMA kernel) + biases
    float sem18[18]; float am2[9];
#pragma unroll
    for (int ch = 0; ch < 18; ++ch)
        sem18[ch] = ws_conv[(size_t)ch * NPIX_ + idx] + sem_b[ch];
#pragma unroll
    for (int k = 0; k < 9; ++k)
        am2[k] = ws_conv[(size_t)(18 + k) * NPIX_ + idx] + m2_b[k];

    float um = 0.0f;
    const float* mb = mask_in + b * HW_;
#pragma unroll
    for (int g = 0; g < G_; ++g) {
#pragma unroll
        for (int k = 0; k < KK_; ++k) {
            const float dy = (g == 0) ? (reg18[2 * k]     + reg_b[2 * k])     : sem18[2 * k];
            const float dx = (g == 0) ? (reg18[2 * k + 1] + reg_b[2 * k + 1]) : sem18[2 * k + 1];
            const float py = dy + (float)y + (float)(k / 3 - 1);
            const float px = dx + (float)x + (float)(k % 3 - 1);
            const int oidx = ((b * G_ + g) * KK_ + k) * HW_ + hw;
            ws_py[oidx] = py;
            ws_px[oidx] = px;
            ws_dm[oidx] = (g == 0) ? sigmoid_(am1[k] + m1_b[k]) : sigmoid_(am2[k]);

            const float y0f = floorf(py), x0f = floorf(px);
            const int   y0  = (int)y0f,   x0  = (int)x0f;
            const float ly = py - y0f, lx = px - x0f;
            um += (1.f - ly) * (1.f - lx) * samp_mask_(mb, y0,     x0);
            um += (1.f - ly) * lx         * samp_mask_(mb, y0,     x0 + 1);
            um += ly         * (1.f - lx) * samp_mask_(mb, y0 + 1, x0);
            um += ly         * lx         * samp_mask_(mb, y0 + 1, x0 + 1);
        }
    }
    um = fminf(fmaxf((float)(2 * CG_) * um, 0.0f), 1.0f);
    ws_um[idx]  = um;
    out_um[idx] = um;   // second output of the reference
}

// ---------------------------------------------------------------------------
// Kernel C: fused bilinear-gather + fp32 WMMA main GEMM.
//   out[o,p] = (sum_k W[o,k] * S[k,p] + bias[o]) * update_mask[p]
//   M = 64 (COUT), K = 576, N = 73728 pixels; 64 pixels per block.
// ---------------------------------------------------------------------------
__global__ void __launch_bounds__(256)
dsc_gemm_kernel(const float* __restrict__ input,  const float* __restrict__ weight,
                const float* __restrict__ bias,
                const float* __restrict__ ws_py,  const float* __restrict__ ws_px,
                const float* __restrict__ ws_dm,  const float* __restrict__ ws_um,
                float* __restrict__ out)
{
    extern __shared__ float smem[];   // S tile: [KTOT_][LDS_STRIDE]
    const int pbase = blockIdx.x * TN_;

    // ---- Phase 1: stage gathered & modulated im2col tile into LDS ----
    for (int task = threadIdx.x; task < G_ * KK_ * TN_; task += 256) {
        const int n  = task & (TN_ - 1);
        const int gk = task >> 6;                 // TN_ == 64
        const int g  = gk / KK_;
        const int k  = gk - g * KK_;
        const int p  = pbase + n;
        const int b  = p / HW_;
        const int hw = p - b * HW_;
        const int oidx = ((b * G_ + g) * KK_ + k) * HW_ + hw;

        const float py = ws_py[oidx];
        const float px = ws_px[oidx];
        const float dm = ws_dm[oidx];

        const float y0f = floorf(py), x0f = floorf(px);
        const int   y0  = (int)y0f,   x0  = (int)x0f;
        const float ly = py - y0f, lx = px - x0f;
        const bool vy0 = (y0 >= 0)     && (y0 < H_);
        const bool vy1 = (y0 + 1 >= 0) && (y0 + 1 < H_);
        const bool vx0 = (x0 >= 0)     && (x0 < W_);
        const bool vx1 = (x0 + 1 >= 0) && (x0 + 1 < W_);
        // fold validity into the corner weights (reference zeroes invalid corners)
        const float w00 = (vy0 && vx0) ? (1.f - ly) * (1.f - lx) * dm : 0.f;
        const float w01 = (vy0 && vx1) ? (1.f - ly) * lx         * dm : 0.f;
        const float w10 = (vy1 && vx0) ? ly         * (1.f - lx) * dm : 0.f;
        const float w11 = (vy1 && vx1) ? ly         * lx         * dm : 0.f;
        const int yc0 = min(max(y0, 0),     H_ - 1);
        const int yc1 = min(max(y0 + 1, 0), H_ - 1);
        const int xc0 = min(max(x0, 0),     W_ - 1);
        const int xc1 = min(max(x0 + 1, 0), W_ - 1);
        const int i00 = yc0 * W_ + xc0, i01 = yc0 * W_ + xc1;
        const int i10 = yc1 * W_ + xc0, i11 = yc1 * W_ + xc1;

        const float* chan = input + (size_t)(b * CIN_ + g * CG_) * HW_;
#pragma unroll 4
        for (int c = 0; c < CG_; ++c) {
            const float v = w00 * chan[i00] + w01 * chan[i01]
                          + w10 * chan[i10] + w11 * chan[i11];
            smem[((g * CG_ + c) * KK_ + k) * LDS_STRIDE + n] = v;
            chan += HW_;
        }
    }
    __syncthreads();

    // ---- Phase 2: fp32 WMMA sweep (uniform EXEC from here on) ----
    const int lane   = threadIdx.x & 31;
    const int wv     = threadIdx.x >> 5;   // 0..7
    const int lane16 = lane & 15;
    const int half   = lane >> 4;          // 0: K=k0,k0+1   1: K=k0+2,k0+3
    const int koff   = half * 2;
    const int nt     = wv & 3;             // N tile (16 pixels)
    const int mt0    = wv >> 2;            // M tiles: wave does mt0 and mt0+2
    const int mt1    = mt0 + 2;

    const float* wrow0 = weight + (size_t)(mt0 * 16 + lane16) * KTOT_;
    const float* wrow1 = weight + (size_t)(mt1 * 16 + lane16) * KTOT_;
    const int bcol = nt * 16 + lane16;

    v8f acc0 = {0.f, 0.f, 0.f, 0.f, 0.f, 0.f, 0.f, 0.f};
    v8f acc1 = {0.f, 0.f, 0.f, 0.f, 0.f, 0.f, 0.f, 0.f};

#pragma unroll 4
    for (int k0 = 0; k0 < KTOT_; k0 += 4) {
        // A fragments per ISA 32-bit A 16x4 layout (lane16 = M, half selects K pair)
        const v2f a0 = *(const v2f*)(wrow0 + k0 + koff);
        const v2f a1 = *(const v2f*)(wrow1 + k0 + koff);
        // B fragment: rows K=k0+koff, k0+koff+1 of the staged S tile
        const float* srow = smem + (k0 + koff) * LDS_STRIDE + bcol;
        const v2f bv = { srow[0], srow[LDS_STRIDE] };
        acc0 = __builtin_amdgcn_wmma_f32_16x16x4_f32(false, a0, false, bv,
                                                     (short)0, acc0, false, false);
        acc1 = __builtin_amdgcn_wmma_f32_16x16x4_f32(false, a1, false, bv,
                                                     (short)0, acc1, false, false);
    }

    // ---- Epilogue: bias + update_mask, C layout: VGPR j -> M = 8*half + j ----
    const int p  = pbase + nt * 16 + lane16;
    const int b  = p / HW_;
    const int hw = p - b * HW_;
    const float umv = ws_um[p];
#pragma unroll
    for (int j = 0; j < 8; ++j) {
        const int o0 = mt0 * 16 + half * 8 + j;
        out[(size_t)(b * COUT_ + o0) * HW_ + hw] = (acc0[j] + bias[o0]) * umv;
        const int o1 = mt1 * 16 + half * 8 + j;
        out[(size_t)(b * COUT_ + o1) * HW_ + hw] = (acc1[j] + bias[o1]) * umv;
    }
}

// ---------------------------------------------------------------------------
extern "C" void kernel_launch(void* const* d_in, const int* in_sizes, int n_in,
                              void* d_out, int out_size, void* d_ws, size_t ws_size,
                              hipStream_t stream)
{
    const float* input   = (const float*)d_in[0];
    const float* mask_in = (const float*)d_in[1];
    const float* weight  = (const float*)d_in[2];
    const float* bias    = (const float*)d_in[3];
    const float* sem_w   = (const float*)d_in[4];
    const float* sem_b   = (const float*)d_in[5];
    const float* reg_w   = (const float*)d_in[6];
    const float* reg_b   = (const float*)d_in[7];
    const float* m1_w    = (const float*)d_in[8];
    const float* m1_b    = (const float*)d_in[9];
    const float* m2_w    = (const float*)d_in[10];
    const float* m2_b    = (const float*)d_in[11];

    float* out = (float*)d_out;
    float* out_um = out + (size_t)B_ * COUT_ * HW_;   // second output region

    const size_t CONVSZ = (size_t)NCHPAD * NPIX_;     // padded 32-channel conv result
    const size_t OFFSZ  = (size_t)B_ * G_ * KK_ * HW_;
    float* ws_conv = (float*)d_ws;
    float* ws_py   = ws_conv + CONVSZ;
    float* ws_px   = ws_py + OFFSZ;
    float* ws_dm   = ws_px + OFFSZ;
    float* ws_um   = ws_dm + OFFSZ;

    const int smem_bytes = KTOT_ * LDS_STRIDE * (int)sizeof(float);  // ~146 KB of 320 KB/WGP

    hipFuncSetAttribute((const void*)dsc_conv27_kernel,
                        hipFuncAttributeMaxDynamicSharedMemorySize, smem_bytes);
    dsc_conv27_kernel<<<NPIX_ / TN_, 256, smem_bytes, stream>>>(
        input, sem_w, m2_w, ws_conv);

    dsc_offsets_kernel<<<(NPIX_ + 255) / 256, 256, 0, stream>>>(
        mask_in, ws_conv, sem_b, reg_w, reg_b, m1_w, m1_b, m2_b,
        ws_py, ws_px, ws_dm, ws_um, out_um);

    hipFuncSetAttribute((const void*)dsc_gemm_kernel,
                        hipFuncAttributeMaxDynamicSharedMemorySize, smem_bytes);
    dsc_gemm_kernel<<<NPIX_ / TN_, 256, smem_bytes, stream>>>(
        input, weight, bias, ws_py, ws_px, ws_dm, ws_um, out);
}